// NetGIN_44186623541948
// MI455X (gfx1250) — compile-verified
//
#include <hip/hip_runtime.h>
#include <math.h>

// ---------------------------------------------------------------------------
// GIN (5 layers, F=D=10, O=10) for MI455X / gfx1250.
// Feature rows padded to stride 12 (48B, 16-aligned) so they load as float4s
// and so K pads to 12 = 3 WMMA f32 16x16x4 K-steps.
// ---------------------------------------------------------------------------

#define FSTRIDE 12

typedef __attribute__((ext_vector_type(2))) float v2f;
typedef __attribute__((ext_vector_type(8))) float v8f;

__global__ __launch_bounds__(256) void zero_f32(float* __restrict__ p, int n) {
    int i = blockIdx.x * blockDim.x + threadIdx.x;
    if (i < n) p[i] = 0.0f;
}

// x[n][0..9] = h[n][0..9]; pads zeroed; cnt[gid[n]] += 1
__global__ __launch_bounds__(256) void init_kernel(const float* __restrict__ h,
                                                   const int* __restrict__ gid,
                                                   float* __restrict__ x,
                                                   float* __restrict__ cnt, int N) {
    int n = blockIdx.x * blockDim.x + threadIdx.x;
    if (n >= N) return;
    const float* hr = h + (size_t)n * 10;
    float* xr = x + (size_t)n * FSTRIDE;
#pragma unroll
    for (int f = 0; f < 10; ++f) xr[f] = hr[f];
    xr[10] = 0.0f; xr[11] = 0.0f;
    atomicAdd(&cnt[gid[n]], 1.0f);
}

// z = x (rows are 3 float4s)
__global__ __launch_bounds__(256) void copyz_kernel(const float4* __restrict__ x4,
                                                    float4* __restrict__ z4, int n4) {
    int i = blockIdx.x * blockDim.x + threadIdx.x;
    if (i < n4) z4[i] = x4[i];
}

// z[dst] += x[src]  (10 scalar f32 atomics per edge; z,x are L2-resident)
__global__ __launch_bounds__(256) void edge_kernel(const int* __restrict__ src,
                                                   const int* __restrict__ dst,
                                                   const float* __restrict__ x,
                                                   float* __restrict__ z, int E) {
    int e = blockIdx.x * blockDim.x + threadIdx.x;
    if (e >= E) return;
    // stream the index arrays ahead of the gather (global_prefetch_b8)
    if (e + 16384 < E) {
        __builtin_prefetch(src + e + 16384, 0, 3);
        __builtin_prefetch(dst + e + 16384, 0, 3);
    }
    int s = src[e], d = dst[e];
    const float* xs = x + (size_t)s * FSTRIDE;
    float* zd = z + (size_t)d * FSTRIDE;
    float4 a = *(const float4*)(xs);
    float4 b = *(const float4*)(xs + 4);
    float2 c = *(const float2*)(xs + 8);
    atomicAdd(zd + 0, a.x); atomicAdd(zd + 1, a.y);
    atomicAdd(zd + 2, a.z); atomicAdd(zd + 3, a.w);
    atomicAdd(zd + 4, b.x); atomicAdd(zd + 5, b.y);
    atomicAdd(zd + 6, b.z); atomicAdd(zd + 7, b.w);
    atomicAdd(zd + 8, c.x); atomicAdd(zd + 9, c.y);
}

// Per-wave 16-node tiles: x = relu(relu(z@W1+b1)@W2+b2) via chained
// V_WMMA_F32_16X16X4_F32 (K padded 10->12 => 3 K-steps per GEMM).
// Also accumulates per-graph sums S[g][d] with atomics.
__global__ __launch_bounds__(256) void mlp_wmma_kernel(
    const float* __restrict__ z, float* __restrict__ x,
    const float* __restrict__ W1, const float* __restrict__ b1,
    const float* __restrict__ W2, const float* __restrict__ b2,
    const int* __restrict__ gid, float* __restrict__ S, int ntiles) {
    __shared__ float lds[8][16 * 17];  // one 16x16 tile per wave, stride 17

    const int lane = threadIdx.x & 31;
    const int wave = threadIdx.x >> 5;
    const int wid = blockIdx.x * (blockDim.x >> 5) + wave;
    const int nwaves = gridDim.x * (blockDim.x >> 5);

    const int n  = lane & 15;  // A-row index / B,C column index
    const int hi = lane >> 4;  // K/M half select

    // B tiles for W1, W2: vgpr v of half h holds row K = 4k + 2h + v, col n.
    v2f B1k[3], B2k[3];
#pragma unroll
    for (int k = 0; k < 3; ++k) {
        int r0 = 4 * k + 2 * hi, r1 = r0 + 1;
        float w10 = (r0 < 10 && n < 10) ? W1[r0 * 10 + n] : 0.0f;
        float w11 = (r1 < 10 && n < 10) ? W1[r1 * 10 + n] : 0.0f;
        float w20 = (r0 < 10 && n < 10) ? W2[r0 * 10 + n] : 0.0f;
        float w21 = (r1 < 10 && n < 10) ? W2[r1 * 10 + n] : 0.0f;
        B1k[k] = (v2f){w10, w11};
        B2k[k] = (v2f){w20, w21};
    }
    const float bb1 = (n < 10) ? b1[n] : 0.0f;
    const float bb2 = (n < 10) ? b2[n] : 0.0f;
    float* my_lds = lds[wave];

    for (int t = wid; t < ntiles; t += nwaves) {
        const int base = t * 16;
        // ---- GEMM 1: H = z_tile @ W1 -------------------------------------
        const float* zr = z + (size_t)(base + n) * FSTRIDE;
        v8f c = {0.f, 0.f, 0.f, 0.f, 0.f, 0.f, 0.f, 0.f};
#pragma unroll
        for (int k = 0; k < 3; ++k) {
            v2f a = *(const v2f*)(zr + 4 * k + 2 * hi);
            c = __builtin_amdgcn_wmma_f32_16x16x4_f32(
                false, a, false, B1k[k], (short)0, c, false, false);
        }
        // bias + relu; element [m][n], m = v + 8*hi
#pragma unroll
        for (int v = 0; v < 8; ++v) {
            float t1 = c[v] + bb1;
            c[v] = t1 > 0.0f ? t1 : 0.0f;
        }
        // ---- transpose C-layout -> A-layout through LDS ------------------
#pragma unroll
        for (int v = 0; v < 8; ++v) my_lds[(v + 8 * hi) * 17 + n] = c[v];
        asm volatile("s_wait_dscnt 0" ::: "memory");
        // ---- GEMM 2: X = H @ W2 ------------------------------------------
        v8f c2 = {0.f, 0.f, 0.f, 0.f, 0.f, 0.f, 0.f, 0.f};
#pragma unroll
        for (int k = 0; k < 3; ++k) {
            int col = 4 * k + 2 * hi;
            v2f a = (v2f){my_lds[n * 17 + col], my_lds[n * 17 + col + 1]};
            c2 = __builtin_amdgcn_wmma_f32_16x16x4_f32(
                false, a, false, B2k[k], (short)0, c2, false, false);
        }
        // bias + relu, store x rows (stride 12; pad cols are exact zeros),
        // and accumulate per-graph sums.
#pragma unroll
        for (int v = 0; v < 8; ++v) {
            float t2 = c2[v] + bb2;
            t2 = t2 > 0.0f ? t2 : 0.0f;
            int m = base + v + 8 * hi;
            if (n < 12) x[(size_t)m * FSTRIDE + n] = t2;
            if (n < 10) atomicAdd(&S[(size_t)gid[m] * FSTRIDE + n], t2);
        }
    }
}

// scalar fallback for a non-multiple-of-16 node tail (unused when N%16==0)
__global__ __launch_bounds__(256) void mlp_tail_kernel(
    const float* __restrict__ z, float* __restrict__ x,
    const float* __restrict__ W1, const float* __restrict__ b1,
    const float* __restrict__ W2, const float* __restrict__ b2,
    const int* __restrict__ gid, float* __restrict__ S, int start, int N) {
    int node = start + blockIdx.x * blockDim.x + threadIdx.x;
    if (node >= N) return;
    const float* zr = z + (size_t)node * FSTRIDE;
    float h1[10];
#pragma unroll
    for (int j = 0; j < 10; ++j) {
        float a = b1[j];
#pragma unroll
        for (int k = 0; k < 10; ++k) a += zr[k] * W1[k * 10 + j];
        h1[j] = a > 0.0f ? a : 0.0f;
    }
    float* xr = x + (size_t)node * FSTRIDE;
    int g = gid[node];
#pragma unroll
    for (int j = 0; j < 10; ++j) {
        float a = b2[j];
#pragma unroll
        for (int k = 0; k < 10; ++k) a += h1[k] * W2[k * 10 + j];
        a = a > 0.0f ? a : 0.0f;
        xr[j] = a;
        atomicAdd(&S[(size_t)g * FSTRIDE + j], a);
    }
    xr[10] = 0.0f; xr[11] = 0.0f;
}

// out[g][o] += (S[g]/cnt[g]) @ L   (G*O = 10000 threads, trivial)
__global__ __launch_bounds__(256) void readout_kernel(const float* __restrict__ S,
                                                      const float* __restrict__ cnt,
                                                      const float* __restrict__ L,
                                                      float* __restrict__ out, int G) {
    int i = blockIdx.x * blockDim.x + threadIdx.x;
    if (i >= G * 10) return;
    int g = i / 10, o = i % 10;
    float inv = 1.0f / cnt[g];
    float acc = 0.0f;
#pragma unroll
    for (int d = 0; d < 10; ++d) acc += S[(size_t)g * FSTRIDE + d] * L[d * 10 + o];
    out[i] += acc * inv;
}

__global__ __launch_bounds__(256) void sigmoid_kernel(float* __restrict__ p, int n) {
    int i = blockIdx.x * blockDim.x + threadIdx.x;
    if (i < n) p[i] = 1.0f / (1.0f + expf(-p[i]));
}

extern "C" void kernel_launch(void* const* d_in, const int* in_sizes, int n_in,
                              void* d_out, int out_size, void* d_ws, size_t ws_size,
                              hipStream_t stream) {
    const float* h   = (const float*)d_in[0];
    const int*   src = (const int*)d_in[1];
    const int*   dst = (const int*)d_in[2];
    const int*   gid = (const int*)d_in[3];
    const float* W1  = (const float*)d_in[4];
    const float* b1  = (const float*)d_in[5];
    const float* W2  = (const float*)d_in[6];
    const float* b2  = (const float*)d_in[7];
    const float* Lm  = (const float*)d_in[8];

    const int N = in_sizes[0] / 10;
    const int E = in_sizes[1];
    const int G = out_size / 10;
    float* out = (float*)d_out;

    // workspace carve-out (everything L2-resident: ~48.1 MB)
    char* ws = (char*)d_ws;
    size_t off = 0;
    auto alignUp = [](size_t v) { return (v + 255) & ~(size_t)255; };
    float* x = (float*)(ws + off); off = alignUp(off + (size_t)N * FSTRIDE * sizeof(float));
    float* z = (float*)(ws + off); off = alignUp(off + (size_t)N * FSTRIDE * sizeof(float));
    float* S = (float*)(ws + off); off = alignUp(off + (size_t)G * FSTRIDE * sizeof(float));
    float* cnt = (float*)(ws + off);

    const int T = 256;
    auto blk = [T](int n) { return (n + T - 1) / T; };

    zero_f32<<<blk(G), T, 0, stream>>>(cnt, G);
    zero_f32<<<blk(G * 10), T, 0, stream>>>(out, G * 10);
    init_kernel<<<blk(N), T, 0, stream>>>(h, gid, x, cnt, N);

    const int ntiles = N / 16;
    const int rem = N - ntiles * 16;
    const int n4 = N * (FSTRIDE / 4);

    for (int i = 0; i < 5; ++i) {
        zero_f32<<<blk(G * FSTRIDE), T, 0, stream>>>(S, G * FSTRIDE);
        copyz_kernel<<<blk(n4), T, 0, stream>>>((const float4*)x, (float4*)z, n4);
        edge_kernel<<<blk(E), T, 0, stream>>>(src, dst, x, z, E);
        if (ntiles > 0)
            mlp_wmma_kernel<<<1024, T, 0, stream>>>(z, x, W1 + i * 100, b1 + i * 10,
                                                    W2 + i * 100, b2 + i * 10, gid, S, ntiles);
        if (rem > 0)
            mlp_tail_kernel<<<blk(rem), T, 0, stream>>>(z, x, W1 + i * 100, b1 + i * 10,
                                                        W2 + i * 100, b2 + i * 10, gid, S,
                                                        ntiles * 16, N);
        readout_kernel<<<blk(G * 10), T, 0, stream>>>(S, cnt, Lm + i * 100, out, G);
    }
    sigmoid_kernel<<<blk(G * 10), T, 0, stream>>>(out, G * 10);
}